// normSage_14250701488884
// MI455X (gfx1250) — compile-verified
//
#include <hip/hip_runtime.h>
#include <hip/hip_bf16.h>

#define IN_F   128
#define POOL_F 96
#define OUT_F  128

typedef __attribute__((ext_vector_type(2))) float v2f;
typedef __attribute__((ext_vector_type(8))) float v8f;

static __device__ __forceinline__ v8f wmma4(v2f a, v2f b, v8f c) {
  // D = A(16x4) * B(4x16) + C(16x16), fp32
  return __builtin_amdgcn_wmma_f32_16x16x4_f32(false, a, false, b, (short)0, c, false, false);
}

// ---------------------------------------------------------------------------
// Kernel 1: h = relu(x @ pool_W^T + pool_b)     [n x 96]
// block = 256 threads (8 waves), computes 64 rows x 96 cols
// Weights staged in LDS as K-pair float2: Wt2[k/2][n] = {W[n][k], W[n][k+1]}
// ---------------------------------------------------------------------------
__global__ __launch_bounds__(256) void pool_gemm_kernel(
    const float* __restrict__ x, const float* __restrict__ pool_W,
    const float* __restrict__ pool_b, float* __restrict__ h, int n)
{
  constexpr int WSTR2 = 112; // float2 row stride: 2*112 % 64 == 32 -> halves bank-disjoint
  constexpr int XSTR  = 132; // 132 % 64 == 4 -> conflict-free float2 A reads
  __shared__ v2f   Wt2[(IN_F / 2) * WSTR2];  // [k2][nfeat]
  __shared__ float Xs[64 * XSTR];            // [r][k]

  const int t = threadIdx.x;
  const int row0 = blockIdx.x * 64;

  // stage pair-transposed weights (coalesced global reads)
  {
    float* Wf = (float*)Wt2;
    for (int i = t; i < POOL_F * IN_F; i += 256) {
      int nf = i >> 7;       // / 128
      int kk = i & 127;
      Wf[(kk >> 1) * (2 * WSTR2) + nf * 2 + (kk & 1)] = pool_W[i];
    }
  }
  // stage 64x128 x tile, float4, clamp OOB rows
  for (int i4 = t; i4 < 64 * (IN_F / 4); i4 += 256) {
    int r  = i4 >> 5;      // / 32
    int c4 = i4 & 31;
    int row = row0 + r; if (row >= n) row = n - 1;
    const float4 v = ((const float4*)(x + (size_t)row * IN_F))[c4];
    float* dst = &Xs[r * XSTR + c4 * 4];
    dst[0] = v.x; dst[1] = v.y; dst[2] = v.z; dst[3] = v.w;
  }
  __syncthreads();

  const int lane  = t & 31;
  const int wave  = t >> 5;
  const int half  = lane >> 4;
  const int l16   = lane & 15;
  const int rbase = (wave >> 1) * 16;     // 4 row tiles
  const int cbase = (wave & 1) * 48;      // 2 col groups x 3 tiles = 96 cols

  v8f acc[3] = {};
  const float* arow = &Xs[(rbase + l16) * XSTR];

  for (int k0 = 0; k0 < IN_F; k0 += 4) {
    const int ka = k0 + 2 * half;
    v2f a = *(const v2f*)(arow + ka);
    const v2f* w = &Wt2[((k0 >> 1) + half) * WSTR2 + cbase + l16];
    #pragma unroll
    for (int ct = 0; ct < 3; ++ct) {
      acc[ct] = wmma4(a, w[ct * 16], acc[ct]);
    }
  }

  #pragma unroll
  for (int ct = 0; ct < 3; ++ct) {
    const int col = cbase + ct * 16 + l16;
    const float bias = pool_b[col];
    #pragma unroll
    for (int r = 0; r < 8; ++r) {
      const int row = row0 + rbase + half * 8 + r;
      if (row < n) {
        float v = acc[ct][r] + bias;
        h[(size_t)row * POOL_F + col] = v > 0.f ? v : 0.f;
      }
    }
  }
}

// ---------------------------------------------------------------------------
// Kernel 2a: zero agg
// ---------------------------------------------------------------------------
__global__ void zero_kernel(float4* __restrict__ p, int n4)
{
  int i = blockIdx.x * blockDim.x + threadIdx.x;
  if (i < n4) p[i] = make_float4(0.f, 0.f, 0.f, 0.f);
}

// ---------------------------------------------------------------------------
// Kernel 2b: agg[row] += edge_val * h[col]^2   (one wave per edge)
// ---------------------------------------------------------------------------
__global__ __launch_bounds__(256) void spmm_kernel(
    const int* __restrict__ erow, const int* __restrict__ ecol,
    const float* __restrict__ eval, const float* __restrict__ h,
    float* __restrict__ agg, int E)
{
  const int wid  = (blockIdx.x * blockDim.x + threadIdx.x) >> 5;  // global wave id = edge
  const int lane = threadIdx.x & 31;
  if (wid >= E) return;
  const int   r = erow[wid];
  const int   c = ecol[wid];
  const float w = eval[wid];
  const float* hsrc = h   + (size_t)c * POOL_F;
  float*       adst = agg + (size_t)r * POOL_F;
  #pragma unroll
  for (int f = lane; f < POOL_F; f += 32) {
    const float hv = hsrc[f];
    atomicAdd(&adst[f], w * hv * hv);
  }
}

// ---------------------------------------------------------------------------
// Kernel 3: out = h @ fc1^T + sqrt+(agg) @ fc2^T + fc1_b + fc2_b   [n x 128]
// block = 256 threads, 64 rows x 128 cols; K=96 for each of two A operands
// ---------------------------------------------------------------------------
__global__ __launch_bounds__(256) void out_gemm_kernel(
    const float* __restrict__ h, const float* __restrict__ agg,
    const float* __restrict__ fc1_W, const float* __restrict__ fc1_b,
    const float* __restrict__ fc2_W, const float* __restrict__ fc2_b,
    float* __restrict__ out, int n)
{
  constexpr int WSTR2 = 144; // float2 row stride: 2*144 % 64 == 32 -> halves bank-disjoint
  constexpr int ASTR  = 100; // conflict-free float2 A reads
  __shared__ v2f   W1t2[(POOL_F / 2) * WSTR2];  // fc1_W pair-transposed: [k2][o]
  __shared__ v2f   W2t2[(POOL_F / 2) * WSTR2];  // fc2_W pair-transposed: [k2][o]
  __shared__ float Hs[64 * ASTR];               // h tile
  __shared__ float Qs[64 * ASTR];               // sqrt+(agg) tile

  const int t = threadIdx.x;
  const int row0 = blockIdx.x * 64;

  {
    float* W1f = (float*)W1t2;
    float* W2f = (float*)W2t2;
    for (int i = t; i < OUT_F * POOL_F; i += 256) {
      int o = i / POOL_F;
      int k = i - o * POOL_F;
      const int dst = (k >> 1) * (2 * WSTR2) + o * 2 + (k & 1);
      W1f[dst] = fc1_W[i];
      W2f[dst] = fc2_W[i];
    }
  }
  for (int i = t; i < 64 * POOL_F; i += 256) {
    int r = i / POOL_F;
    int k = i - r * POOL_F;
    int row = row0 + r; if (row >= n) row = n - 1;
    const size_t idx = (size_t)row * POOL_F + k;
    Hs[r * ASTR + k] = h[idx];
    const float a = agg[idx];
    Qs[r * ASTR + k] = a > 0.f ? sqrtf(a) : 0.f;
  }
  __syncthreads();

  const int lane  = t & 31;
  const int wave  = t >> 5;
  const int half  = lane >> 4;
  const int l16   = lane & 15;
  const int rbase = (wave >> 1) * 16;   // 4 row tiles
  const int cbase = (wave & 1) * 64;    // 2 col groups x 4 tiles = 128 cols

  v8f acc[4] = {};
  const float* hrow = &Hs[(rbase + l16) * ASTR];
  const float* qrow = &Qs[(rbase + l16) * ASTR];

  for (int k0 = 0; k0 < POOL_F; k0 += 4) {
    const int ka = k0 + 2 * half;
    v2f a1 = *(const v2f*)(hrow + ka);
    v2f a2 = *(const v2f*)(qrow + ka);
    const int wrow = ((k0 >> 1) + half) * WSTR2 + cbase + l16;
    const v2f* w1 = &W1t2[wrow];
    const v2f* w2 = &W2t2[wrow];
    #pragma unroll
    for (int ct = 0; ct < 4; ++ct) {
      acc[ct] = wmma4(a1, w1[ct * 16], acc[ct]);
      acc[ct] = wmma4(a2, w2[ct * 16], acc[ct]);
    }
  }

  #pragma unroll
  for (int ct = 0; ct < 4; ++ct) {
    const int col = cbase + ct * 16 + l16;
    const float bias = fc1_b[col] + fc2_b[col];
    #pragma unroll
    for (int r = 0; r < 8; ++r) {
      const int row = row0 + rbase + half * 8 + r;
      if (row < n)
        out[(size_t)row * OUT_F + col] = acc[ct][r] + bias;
    }
  }
}

// ---------------------------------------------------------------------------
extern "C" void kernel_launch(void* const* d_in, const int* in_sizes, int n_in,
                              void* d_out, int out_size, void* d_ws, size_t ws_size,
                              hipStream_t stream)
{
  const float* x      = (const float*)d_in[0];
  const int*   erow   = (const int*)  d_in[1];
  const int*   ecol   = (const int*)  d_in[2];
  const float* eval   = (const float*)d_in[3];
  const float* pool_W = (const float*)d_in[4];
  const float* pool_b = (const float*)d_in[5];
  const float* fc1_W  = (const float*)d_in[6];
  const float* fc1_b  = (const float*)d_in[7];
  const float* fc2_W  = (const float*)d_in[8];
  const float* fc2_b  = (const float*)d_in[9];
  float* out = (float*)d_out;

  const int n = in_sizes[0] / IN_F;   // 50000
  const int E = in_sizes[1];          // 800000

  float* h   = (float*)d_ws;                  // n * 96 floats
  float* agg = h + (size_t)n * POOL_F;        // n * 96 floats

  // 1) h = relu(x @ pool_W^T + b)
  {
    dim3 grid((n + 63) / 64);
    pool_gemm_kernel<<<grid, 256, 0, stream>>>(x, pool_W, pool_b, h, n);
  }
  // 2a) agg = 0
  {
    const int n4 = (n * POOL_F) / 4;
    dim3 grid((n4 + 255) / 256);
    zero_kernel<<<grid, 256, 0, stream>>>((float4*)agg, n4);
  }
  // 2b) agg[row] += w * h[col]^2
  {
    const long long waves = E;              // one wave per edge
    dim3 grid((unsigned)((waves * 32 + 255) / 256));
    spmm_kernel<<<grid, 256, 0, stream>>>(erow, ecol, eval, h, agg, E);
  }
  // 3) out = h @ fc1^T + sqrt+(agg) @ fc2^T + b1 + b2
  {
    dim3 grid((n + 63) / 64);
    out_gemm_kernel<<<grid, 256, 0, stream>>>(h, agg, fc1_W, fc1_b, fc2_W, fc2_b, out, n);
  }
}